// YOLOX_85212151152869
// MI455X (gfx1250) — compile-verified
//
#include <hip/hip_runtime.h>
#include <hip/hip_bf16.h>
#include <stdint.h>

// ---------------------------------------------------------------------------
// YOLOX head + top-k + NMS for MI455X (gfx1250, wave32, WMMA)
// ---------------------------------------------------------------------------

typedef float v2f  __attribute__((ext_vector_type(2)));
typedef float v8f  __attribute__((ext_vector_type(8)));
typedef unsigned int u32x4 __attribute__((ext_vector_type(4)));
typedef int   i32x4 __attribute__((ext_vector_type(4)));
typedef int   i32x8 __attribute__((ext_vector_type(8)));

#define KDIM 256     // input channels
#define NCH  96      // 85 output channels padded to 6 WMMA N-tiles
#define NTILES_N 6
#define TOTAL_PIX 33600
#define TOPK_N 1000
#define NMS_N 3000
#define CONF_TH 0.05f
#define NMS_TH 0.6f
#define IMG_MAX 1280.0f
#define CLS_OFF 8192.0f

__device__ __forceinline__ float fsigmoid(float x) {
  return 1.0f / (1.0f + __expf(-x));
}

// ---------------------------------------------------------------------------
// 1) Pack per-level weights into a single padded 96x256 matrix + 96 biases.
//    Channel order: [0..79]=cls, [80]=obj, [81..84]=reg, [85..95]=zero pad.
// ---------------------------------------------------------------------------
__global__ void pack_kernel(const float* __restrict__ Wobj,
                            const float* __restrict__ bobj,
                            const float* __restrict__ Wcls,
                            const float* __restrict__ bcls,
                            const float* __restrict__ Wreg,
                            const float* __restrict__ breg,
                            float* __restrict__ Wl,
                            float* __restrict__ Bl) {
  int idx = blockIdx.x * blockDim.x + threadIdx.x;
  if (idx >= NCH * KDIM) return;
  int ch = idx >> 8;
  int k  = idx & 255;
  float w, b;
  if (ch < 80)      { w = Wcls[ch * KDIM + k];        b = bcls[ch]; }
  else if (ch == 80){ w = Wobj[k];                    b = bobj[0]; }
  else if (ch < 85) { w = Wreg[(ch - 81) * KDIM + k]; b = breg[ch - 81]; }
  else              { w = 0.0f;                       b = 0.0f; }
  Wl[idx] = w;
  if (k == 0) Bl[ch] = b;
}

// ---------------------------------------------------------------------------
// 2) Head GEMM per level:  out[p, n] = sum_k feat[k, p] * W[n, k]  (+bias)
//    One wave32 per 16-pixel tile, 6 N-tiles of 16 channels, K=256 in steps
//    of 4 via V_WMMA_F32_16X16X4_F32. Weights staged to LDS with the TDM.
// ---------------------------------------------------------------------------
__global__ __launch_bounds__(256)
void head_kernel(const float* __restrict__ feat,
                 const float* __restrict__ Wl,
                 const float* __restrict__ Bl,
                 float* __restrict__ scores,
                 int*   __restrict__ labels,
                 float* __restrict__ boxes,
                 int HW, int Wimg, float stride, int pixBase, int numTiles) {
  __shared__ float lds_w[NCH * KDIM];        // 96 KB: level weight matrix
  __shared__ float lds_out[8][16 * NCH];     // 48 KB: per-wave result staging

  const int tid  = threadIdx.x;
  const int wave = tid >> 5;
  const int lane = tid & 31;

  // ---- Stage weights (96*256 f32, contiguous) into LDS ----
#if defined(__gfx1250__) && __has_builtin(__builtin_amdgcn_tensor_load_to_lds)
  if (tid == 0) {
    unsigned long long ga = (unsigned long long)(uintptr_t)Wl;
    unsigned int ldsAddr  = (unsigned int)(uintptr_t)&lds_w[0]; // low 32b = LDS offset
    const unsigned nelem  = NCH * KDIM;                         // 24576 f32 elems
    u32x4 g0;
    g0[0] = 1u;                                       // count=1, user descriptor
    g0[1] = ldsAddr;                                  // lds_addr (bytes)
    g0[2] = (unsigned)(ga & 0xFFFFFFFFu);             // global_addr[31:0]
    g0[3] = ((unsigned)(ga >> 32) & 0x01FFFFFFu)      // global_addr[56:32]
            | (2u << 30);                             // type = 2 ("image")
    i32x8 g1 = {0, 0, 0, 0, 0, 0, 0, 0};
    g1[0] = (int)(2u << 16);                          // data_size = 4 bytes
    g1[1] = (int)((nelem & 0xFFFFu) << 16);           // tensor_dim0[15:0]
    g1[2] = (int)(((nelem >> 16) & 0xFFFFu)           // tensor_dim0[31:16]
            | (1u << 16));                            // tensor_dim1 = 1
    g1[3] = (int)((nelem & 0xFFFFu) << 16);           // tile_dim0 = nelem (1-D tile)
    g1[5] = (int)nelem;                               // tensor_dim0_stride
    i32x4 gz = {0, 0, 0, 0};
    i32x8 gz8 = {0, 0, 0, 0, 0, 0, 0, 0};
    __builtin_amdgcn_tensor_load_to_lds(g0, g1, gz, gz, gz8, 0);
    __builtin_amdgcn_s_wait_tensorcnt(0);
  }
#else
  for (int i = tid; i < NCH * KDIM; i += 256) lds_w[i] = Wl[i];
#endif
  __syncthreads();

  const int tile = blockIdx.x * 8 + wave;
  if (tile < numTiles) {
    const int p0    = tile * 16;
    const int halfk = (lane >> 4) * 2;   // 0 (lanes 0-15) / 2 (lanes 16-31)
    const int m     = lane & 15;
    const float* fbase = feat + p0 + m;

    v8f acc[NTILES_N];
    v8f zero = {0.f, 0.f, 0.f, 0.f, 0.f, 0.f, 0.f, 0.f};
#pragma unroll
    for (int t = 0; t < NTILES_N; t++) acc[t] = zero;

#pragma unroll 4
    for (int k0 = 0; k0 < KDIM; k0 += 4) {
      // A: 16x4 f32 tile — VGPR0 holds K=k0 / k0+2, VGPR1 holds K=k0+1 / k0+3
      v2f a;
      a.x = fbase[(k0 + halfk + 0) * HW];
      a.y = fbase[(k0 + halfk + 1) * HW];
      if (k0 + 16 < KDIM)
        __builtin_prefetch(fbase + (k0 + 16) * HW, 0, 1);  // global_prefetch
#pragma unroll
      for (int t = 0; t < NTILES_N; t++) {
        // B: 4x16 f32 tile from LDS, mirrored striping
        const int wrow = (t * 16 + m) * KDIM + halfk;
        v2f b;
        b.x = lds_w[wrow + k0 + 0];
        b.y = lds_w[wrow + k0 + 1];
        acc[t] = __builtin_amdgcn_wmma_f32_16x16x4_f32(
            false, a, false, b, (short)0, acc[t], false, false);
      }
    }

    // Spill D tiles to LDS: VGPR r -> pixel (r + 8*(lane>=16)), channel lane%16
    const int rowBase = (lane >> 4) * 8;
#pragma unroll
    for (int t = 0; t < NTILES_N; t++)
#pragma unroll
      for (int r = 0; r < 8; r++)
        lds_out[wave][(rowBase + r) * NCH + t * 16 + m] = acc[t][r];

    // Per-pixel post-processing (lanes 0..15, one pixel each)
    if (lane < 16) {
      const float* v = &lds_out[wave][lane * NCH];
      const int p = p0 + lane;
      const float so = fsigmoid(v[80] + Bl[80]);
      float best = -1.0f;
      int bi = 0;
#pragma unroll 8
      for (int c = 0; c < 80; c++) {
        float pr = so * fsigmoid(v[c] + Bl[c]);
        if (pr > best) { best = pr; bi = c; }
      }
      const float r0 = __expf(v[81] + Bl[81]) * stride;
      const float r1 = __expf(v[82] + Bl[82]) * stride;
      const float r2 = __expf(v[83] + Bl[83]) * stride;
      const float r3 = __expf(v[84] + Bl[84]) * stride;
      const int ax = p % Wimg;
      const int ay = p / Wimg;
      const float cx = ((float)ax + 0.5f) * stride;
      const float cy = ((float)ay + 0.5f) * stride;
      const int gp = pixBase + p;
      scores[gp] = best;
      labels[gp] = bi;
      boxes[gp * 4 + 0] = cx - r0;
      boxes[gp * 4 + 1] = cy - r1;
      boxes[gp * 4 + 2] = cx + r2;
      boxes[gp * 4 + 3] = cy + r3;
    }
  }
}

// monotonic key: larger float <-> larger unsigned
__device__ __forceinline__ unsigned fkey(float s) {
  unsigned u = __float_as_uint(s);
  return (u & 0x80000000u) ? ~u : (u | 0x80000000u);
}

// ---------------------------------------------------------------------------
// 3) Per-level exact top-1000: radix-histogram threshold + bitonic sort.
// ---------------------------------------------------------------------------
__global__ __launch_bounds__(1024)
void topk_kernel(const float* __restrict__ scores,
                 const int*   __restrict__ labels,
                 const float* __restrict__ boxes,
                 float* __restrict__ tks, int* __restrict__ tkl,
                 float* __restrict__ tkb, float* __restrict__ out,
                 int n, int base, int lvl) {
  __shared__ unsigned hist[2048];
  __shared__ unsigned s_cut, s_need, s_take, s_cutTaken;
  __shared__ unsigned skeys[1024];
  __shared__ unsigned sidx[1024];
  const int tid = threadIdx.x;

  for (int i = tid; i < 2048; i += 1024) hist[i] = 0u;
  if (tid == 0) { s_take = 0u; s_cutTaken = 0u; }
  __syncthreads();

  for (int i = tid; i < n; i += 1024)
    atomicAdd(&hist[fkey(scores[base + i]) >> 21], 1u);
  __syncthreads();

  if (tid == 0) {
    unsigned cum = 0;
    int b = 2047;
    for (; b >= 0; b--) {
      if (cum + hist[b] >= (unsigned)TOPK_N) break;
      cum += hist[b];
    }
    s_cut = (unsigned)b;
    s_need = (unsigned)TOPK_N - cum;
  }
  __syncthreads();
  const unsigned cut = s_cut, need = s_need;

  for (int i = tid; i < n; i += 1024) {
    unsigned key = fkey(scores[base + i]);
    unsigned b = key >> 21;
    unsigned pos = 0xFFFFFFFFu;
    if (b > cut) {
      pos = atomicAdd(&s_take, 1u);
    } else if (b == cut) {
      unsigned q = atomicAdd(&s_cutTaken, 1u);
      if (q < need) pos = atomicAdd(&s_take, 1u);
    }
    if (pos != 0xFFFFFFFFu) { skeys[pos] = key; sidx[pos] = (unsigned)i; }
  }
  __syncthreads();
  if (tid >= (int)s_take) { skeys[tid] = 0u; sidx[tid] = 0u; }

  // bitonic sort, descending, 1024 elements / 1024 threads
  for (unsigned sz = 2; sz <= 1024; sz <<= 1) {
    for (unsigned st = sz >> 1; st > 0; st >>= 1) {
      __syncthreads();
      unsigned i = (unsigned)tid, j = i ^ st;
      if (j > i) {
        bool desc = ((i & sz) == 0);
        unsigned ki = skeys[i], kj = skeys[j];
        if (desc ? (ki < kj) : (ki > kj)) {
          skeys[i] = kj; skeys[j] = ki;
          unsigned t1 = sidx[i]; sidx[i] = sidx[j]; sidx[j] = t1;
        }
      }
    }
  }
  __syncthreads();

  if (tid < TOPK_N) {
    const int oi = lvl * TOPK_N + tid;
    const int src = base + (int)sidx[tid];
    const float sc = scores[src];
    const int lb = labels[src];
    tks[oi] = sc;
    tkl[oi] = lb;
    out[NMS_N * 4 + oi] = sc;                 // scores section
    ((int*)out)[NMS_N * 5 + oi] = lb;         // labels section (int32 bits)
#pragma unroll
    for (int c = 0; c < 4; c++) {
      float bx = boxes[src * 4 + c];
      tkb[oi * 4 + c] = bx;
      float nb = bx * (1.0f / IMG_MAX);
      out[oi * 4 + c] = fminf(fmaxf(nb, 0.0f), 1.0f);  // bboxes section
    }
  }
}

// ---------------------------------------------------------------------------
// 4) Greedy class-offset NMS over the 3000 selected boxes (one workgroup,
//    everything resident in LDS).
// ---------------------------------------------------------------------------
__global__ __launch_bounds__(1024)
void nms_kernel(const float* __restrict__ tks,
                const int*   __restrict__ tkl,
                const float* __restrict__ tkb,
                float* __restrict__ out) {
  __shared__ unsigned okey[4096];
  __shared__ unsigned oidx[4096];
  __shared__ float bx1[NMS_N], by1[NMS_N], bx2[NMS_N], by2[NMS_N], area[NMS_N];
  __shared__ unsigned keep[NMS_N];
  const int tid = threadIdx.x;

  for (int i = tid; i < 4096; i += 1024) {
    if (i < NMS_N) { okey[i] = fkey(tks[i]); oidx[i] = (unsigned)i; }
    else           { okey[i] = 0u;           oidx[i] = 0u; }
  }
  __syncthreads();

  // bitonic argsort, descending, 4096 elements / 1024 threads (x4)
  for (unsigned sz = 2; sz <= 4096; sz <<= 1) {
    for (unsigned st = sz >> 1; st > 0; st >>= 1) {
      __syncthreads();
      for (int m = 0; m < 4; m++) {
        unsigned i = (unsigned)tid + (unsigned)m * 1024u;
        unsigned j = i ^ st;
        if (j > i) {
          bool desc = ((i & sz) == 0);
          unsigned ki = okey[i], kj = okey[j];
          if (desc ? (ki < kj) : (ki > kj)) {
            okey[i] = kj; okey[j] = ki;
            unsigned t1 = oidx[i]; oidx[i] = oidx[j]; oidx[j] = t1;
          }
        }
      }
    }
  }
  __syncthreads();

  // gather class-offset boxes in sorted order
  for (int i = tid; i < NMS_N; i += 1024) {
    const unsigned si = oidx[i];
    const float off = (float)tkl[si] * CLS_OFF;
    const float x1 = tkb[si * 4 + 0] + off;
    const float y1 = tkb[si * 4 + 1] + off;
    const float x2 = tkb[si * 4 + 2] + off;
    const float y2 = tkb[si * 4 + 3] + off;
    bx1[i] = x1; by1[i] = y1; bx2[i] = x2; by2[i] = y2;
    area[i] = (x2 - x1) * (y2 - y1);
    keep[i] = (tks[si] >= CONF_TH) ? 1u : 0u;
  }
  __syncthreads();

  // sequential greedy suppression (reference semantics)
  for (int i = 0; i < NMS_N; i++) {
    if (keep[i]) {
      const float x1 = bx1[i], y1 = by1[i], x2 = bx2[i], y2 = by2[i];
      const float ai = area[i];
      for (int j = i + 1 + tid; j < NMS_N; j += 1024) {
        if (keep[j]) {
          const float xx1 = fmaxf(x1, bx1[j]);
          const float yy1 = fmaxf(y1, by1[j]);
          const float xx2 = fminf(x2, bx2[j]);
          const float yy2 = fminf(y2, by2[j]);
          const float w = fmaxf(1e-10f, xx2 - xx1);
          const float h = fmaxf(1e-10f, yy2 - yy1);
          const float inter = w * h;
          const float iou = inter / (ai + area[j] - inter + 1e-14f);
          if (iou > NMS_TH) keep[j] = 0u;
        }
      }
    }
    __syncthreads();
  }

  // scatter keep back to pre-sort (concatenated top-k) order
  for (int i = tid; i < NMS_N; i += 1024)
    out[NMS_N * 6 + (int)oidx[i]] = keep[i] ? 1.0f : 0.0f;  // keep section
}

// ---------------------------------------------------------------------------
// Host launcher
// ---------------------------------------------------------------------------
extern "C" void kernel_launch(void* const* d_in, const int* in_sizes, int n_in,
                              void* d_out, int out_size, void* d_ws, size_t ws_size,
                              hipStream_t stream) {
  (void)in_sizes; (void)n_in; (void)out_size; (void)ws_size;

  float* ws   = (float*)d_ws;
  float* Wall = ws;                                   // 3 * 96 * 256
  float* Ball = Wall + 3 * NCH * KDIM;                // 3 * 96
  float* sc   = Ball + 3 * NCH;                       // 33600
  int*   lb   = (int*)(sc + TOTAL_PIX);               // 33600
  float* bx   = (float*)(lb + TOTAL_PIX);             // 33600 * 4
  float* tks  = bx + TOTAL_PIX * 4;                   // 3000
  int*   tkl  = (int*)(tks + NMS_N);                  // 3000
  float* tkb  = (float*)(tkl + NMS_N);                // 3000 * 4
  float* out  = (float*)d_out;

  static const int   HWs[3]     = {25600, 6400, 1600};
  static const int   Wimg[3]    = {160, 80, 40};
  static const float strides[3] = {8.0f, 16.0f, 32.0f};
  static const int   bases[3]   = {0, 25600, 32000};

  for (int l = 0; l < 3; l++) {
    const float* Wobj = (const float*)d_in[l * 7 + 1];
    const float* bobj = (const float*)d_in[l * 7 + 2];
    const float* Wcls = (const float*)d_in[l * 7 + 3];
    const float* bcls = (const float*)d_in[l * 7 + 4];
    const float* Wreg = (const float*)d_in[l * 7 + 5];
    const float* breg = (const float*)d_in[l * 7 + 6];
    pack_kernel<<<dim3((NCH * KDIM + 255) / 256), dim3(256), 0, stream>>>(
        Wobj, bobj, Wcls, bcls, Wreg, breg,
        Wall + l * NCH * KDIM, Ball + l * NCH);
  }

  for (int l = 0; l < 3; l++) {
    const float* feat = (const float*)d_in[l * 7 + 0];
    const int numTiles = HWs[l] / 16;
    const int blocks = (numTiles + 7) / 8;
    head_kernel<<<dim3(blocks), dim3(256), 0, stream>>>(
        feat, Wall + l * NCH * KDIM, Ball + l * NCH,
        sc, lb, bx, HWs[l], Wimg[l], strides[l], bases[l], numTiles);
  }

  for (int l = 0; l < 3; l++) {
    topk_kernel<<<dim3(1), dim3(1024), 0, stream>>>(
        sc, lb, bx, tks, tkl, tkb, out, HWs[l], bases[l], l);
  }

  nms_kernel<<<dim3(1), dim3(1024), 0, stream>>>(tks, tkl, tkb, out);
}